// MedianFilter_54125177864460
// MI455X (gfx1250) — compile-verified
//
#include <hip/hip_runtime.h>

#define IMG_W 1024
#define IMG_H 1024
#define TILE_W 256            // output cols per block
#define TILE_H 16             // output rows per block
#define ROWS   4              // output rows per thread  (blockDim.y = TILE_H/ROWS)
#define LROWS  (TILE_H + 2)   // 18 input rows staged in LDS
#define LSTRIDE 264           // floats per LDS row: 1056 B, 16B-aligned, bank-skewed
// LDS col layout per row: [3] = left halo (x-1), [4..259] = x0..x0+255, [260] = right halo

#if defined(__has_builtin)
#if __has_builtin(__builtin_amdgcn_global_load_async_to_lds_b128) && \
    __has_builtin(__builtin_amdgcn_global_load_async_to_lds_b32)
#define USE_ASYNC 1
#else
#define USE_ASYNC 0
#endif
#else
#define USE_ASYNC 0
#endif

#if USE_ASYNC
// Exact parameter types per hipcc diagnostics:
//   b128: (v4i __device__*, v4i __shared__*, imm, imm)
//   b32 : (int __device__*, int __shared__*, imm, imm)
typedef int v4i_t __attribute__((vector_size(4 * sizeof(int))));
typedef __attribute__((address_space(1))) v4i_t* glb_v4i_p;
typedef __attribute__((address_space(3))) v4i_t* lds_v4i_p;
typedef __attribute__((address_space(1))) int*   glb_i32_p;
typedef __attribute__((address_space(3))) int*   lds_i32_p;
#endif

__device__ __forceinline__ float min3f(float a, float b, float c) {
    return fminf(fminf(a, b), c);            // -> v_min3_num_f32
}
__device__ __forceinline__ float max3f(float a, float b, float c) {
    return fmaxf(fmaxf(a, b), c);            // -> v_max3_num_f32
}
__device__ __forceinline__ float med3f(float a, float b, float c) {
    return __builtin_amdgcn_fmed3f(a, b, c); // -> v_med3_num_f32
}

__device__ __forceinline__ int reflect_row(int iy, int H) {
    return iy < 0 ? -iy : (iy >= H ? 2 * H - 2 - iy : iy);
}

__global__ void median3x3_kernel(const float* __restrict__ in,
                                 float* __restrict__ out) {
    __shared__ __align__(16) float tile[LROWS * LSTRIDE];

    const int W = IMG_W, H = IMG_H;
    const int img = blockIdx.z;
    const int X0  = blockIdx.x * TILE_W;
    const int Y0  = blockIdx.y * TILE_H;

    const float* __restrict__ p = in  + (size_t)img * H * W;
    float*       __restrict__ q = out + (size_t)img * H * W;

    const int tx  = threadIdx.x;            // 0..63 -> 4 cols each
    const int ty  = threadIdx.y;            // 0..3  -> 4 rows each
    const int tid = ty * 64 + tx;           // 0..255

    // Reflected halo columns (mirror, edge excluded).
    const int xl = (X0 == 0)           ? 1     : X0 - 1;
    const int xr = (X0 + TILE_W >= W)  ? W - 2 : X0 + TILE_W;

    // ---- Stage 18x258 input tile into LDS (gfx1250 async-to-LDS path) ----
    // Main body: LROWS rows x 64 float4 groups = 1152 b128 transfers.
    for (int i = tid; i < LROWS * 64; i += 256) {
        const int r  = i >> 6;
        const int c  = i & 63;
        const int ry = reflect_row(Y0 - 1 + r, H);
        const float* g = p + (size_t)ry * W + X0 + 4 * c;
        float*       l = &tile[r * LSTRIDE + 4 + 4 * c];   // byte off 1056*r+16c+16 (16B aligned)
#if USE_ASYNC
        __builtin_amdgcn_global_load_async_to_lds_b128(
            (glb_v4i_p)(void*)g, (lds_v4i_p)(void*)l, 0, 0);
#else
        *reinterpret_cast<float4*>(l) = *reinterpret_cast<const float4*>(g);
#endif
    }
    // Halo columns: 2 * LROWS = 36 single floats with reflect-mapped addresses.
    for (int i = tid; i < 2 * LROWS; i += 256) {
        const bool right = (i >= LROWS);
        const int  r  = right ? i - LROWS : i;
        const int  ry = reflect_row(Y0 - 1 + r, H);
        const int  gx = right ? xr : xl;
        const int  lc = right ? (4 + TILE_W) : 3;
        const float* g = p + (size_t)ry * W + gx;
        float*       l = &tile[r * LSTRIDE + lc];
#if USE_ASYNC
        __builtin_amdgcn_global_load_async_to_lds_b32(
            (glb_i32_p)(void*)g, (lds_i32_p)(void*)l, 0, 0);
#else
        *l = *g;
#endif
    }

#if USE_ASYNC
#if defined(__has_builtin) && __has_builtin(__builtin_amdgcn_s_wait_asynccnt)
    __builtin_amdgcn_s_wait_asynccnt(0);
#else
    asm volatile("s_wait_asynccnt 0" ::: "memory");
#endif
#endif
    __syncthreads();

    // Prefetch first row of next vertical tile (gfx1250 global_prefetch_b8).
    {
        int pf = Y0 + TILE_H + 1;
        if (pf >= H) pf = H - 2;
        __builtin_prefetch(p + (size_t)pf * W + X0 + 4 * tx, 0, 1);
    }

    // ---- Compute: rolling horizontal row-sort triples, median-of-9 ----
    const int lbase = 4 * tx;   // this thread's LDS col-group base
    float lo[3][4], mi[3][4], hi[3][4];

#define SORT_ROW(slot, lr_)                                                \
    {                                                                      \
        const float* lr = &tile[(lr_) * LSTRIDE + lbase];                  \
        float  v0 = lr[3];                                                 \
        float4 cc = *reinterpret_cast<const float4*>(lr + 4);              \
        float  v5 = lr[8];                                                 \
        float v1 = cc.x, v2 = cc.y, v3 = cc.z, v4 = cc.w;                  \
        lo[slot][0] = min3f(v0, v1, v2);                                   \
        mi[slot][0] = med3f(v0, v1, v2);                                   \
        hi[slot][0] = max3f(v0, v1, v2);                                   \
        lo[slot][1] = min3f(v1, v2, v3);                                   \
        mi[slot][1] = med3f(v1, v2, v3);                                   \
        hi[slot][1] = max3f(v1, v2, v3);                                   \
        lo[slot][2] = min3f(v2, v3, v4);                                   \
        mi[slot][2] = med3f(v2, v3, v4);                                   \
        hi[slot][2] = max3f(v2, v3, v4);                                   \
        lo[slot][3] = min3f(v3, v4, v5);                                   \
        mi[slot][3] = med3f(v3, v4, v5);                                   \
        hi[slot][3] = max3f(v3, v4, v5);                                   \
    }

    // Output row Y0+k uses LDS rows k..k+2 (LDS row 0 holds input row Y0-1).
    const int k0 = 4 * ty;
    SORT_ROW(0, k0 + 0)
    SORT_ROW(1, k0 + 1)

#pragma unroll
    for (int rr = 0; rr < ROWS; ++rr) {
        const int s2 = (rr + 2) % 3;
        SORT_ROW(s2, k0 + rr + 2)
        const int s0 = rr % 3;
        const int s1 = (rr + 1) % 3;

        float res[4];
#pragma unroll
        for (int j = 0; j < 4; ++j) {
            float mx = max3f(lo[s0][j], lo[s1][j], lo[s2][j]);
            float md = med3f(mi[s0][j], mi[s1][j], mi[s2][j]);
            float mn = min3f(hi[s0][j], hi[s1][j], hi[s2][j]);
            res[j] = med3f(mx, md, mn);
        }
        *reinterpret_cast<float4*>(q + (size_t)(Y0 + k0 + rr) * W + X0 + 4 * tx) =
            make_float4(res[0], res[1], res[2], res[3]);
    }
#undef SORT_ROW
}

extern "C" void kernel_launch(void* const* d_in, const int* in_sizes, int n_in,
                              void* d_out, int out_size, void* d_ws, size_t ws_size,
                              hipStream_t stream) {
    (void)in_sizes; (void)n_in; (void)d_ws; (void)ws_size; (void)out_size;
    const float* x = (const float*)d_in[0];
    float* out = (float*)d_out;

    dim3 block(64, 4, 1);                    // 256 threads = 8 wave32
    dim3 grid(IMG_W / TILE_W,                // 4
              IMG_H / TILE_H,                // 64
              12);                           // B*C images
    median3x3_kernel<<<grid, block, 0, stream>>>(x, out);
}